// TokenEncoder_44212393345441
// MI455X (gfx1250) — compile-verified
//
#include <hip/hip_runtime.h>
#include <hip/hip_bf16.h>

typedef __attribute__((ext_vector_type(2))) float v2f;
typedef __attribute__((ext_vector_type(4))) float v4f;
typedef __attribute__((ext_vector_type(8))) float v8f;

#define B_DIM       4
#define L_DIM       1024
#define NTOK        (B_DIM * L_DIM)
#define LP1         (L_DIM + 1)
#define D_MODEL     768
#define D_IN        32
#define NUM_SIGNALS 128
#define NUM_ROLES   3
#define NUM_KEYS    (NUM_SIGNALS * NUM_ROLES)
#define N_TILES     (D_MODEL / 16)   // 48

// ---------------- bucketing: group tokens by (sid*3 + role), mask-filtered ----

__global__ void te_zero(int* __restrict__ counts) {
    int i = blockIdx.x * blockDim.x + threadIdx.x;
    if (i < NUM_KEYS) counts[i] = 0;
}

__global__ void te_count(const int* __restrict__ sid, const int* __restrict__ role,
                         const unsigned char* __restrict__ mask, int* __restrict__ counts) {
    int t = blockIdx.x * blockDim.x + threadIdx.x;
    if (t < NTOK && mask[t]) {
        atomicAdd(&counts[sid[t] * NUM_ROLES + role[t]], 1);
    }
}

// Single-block LDS Hillis-Steele scan (9 steps) instead of 384 serial
// dependent global round-trips.
__global__ __launch_bounds__(512)
void te_scan(int* __restrict__ counts, int* __restrict__ offsets) {
    __shared__ int s[512];
    int i = threadIdx.x;
    int v = (i < NUM_KEYS) ? counts[i] : 0;
    s[i] = v;
    __syncthreads();
#pragma unroll
    for (int d = 1; d < 512; d <<= 1) {
        int tv = (i >= d) ? s[i - d] : 0;
        __syncthreads();
        s[i] += tv;
        __syncthreads();
    }
    if (i < NUM_KEYS) {
        offsets[i] = s[i] - v;      // exclusive prefix
        counts[i] = 0;              // reset for the scatter pass
    }
    if (i == NUM_KEYS - 1) offsets[NUM_KEYS] = s[i];
}

__global__ void te_scatter(const int* __restrict__ sid, const int* __restrict__ role,
                           const unsigned char* __restrict__ mask,
                           int* __restrict__ counts, const int* __restrict__ offsets,
                           int* __restrict__ list) {
    int t = blockIdx.x * blockDim.x + threadIdx.x;
    if (t < NTOK && mask[t]) {
        int k = sid[t] * NUM_ROLES + role[t];
        int p = atomicAdd(&counts[k], 1);
        list[offsets[k] + p] = t;
    }
}

// ---------------- per-key GEMM via V_WMMA_F32_16X16X4_F32 (exact f32 path) ----
// One block per key; 4 waves split the 48 N-tiles; loop over M-tiles of 16
// bucketed tokens; K=32 handled as 8 chained K=4 f32 WMMAs.
// Store path is branch-free: invalid D rows write into a scratch dump region,
// so EXEC stays all-1s and no list[] reloads happen inside the N loop.

__global__ __launch_bounds__(128)
void te_wmma_gemm(const float* __restrict__ emb, const float* __restrict__ W,
                  const float* __restrict__ bias, const int* __restrict__ offsets,
                  const int* __restrict__ list, float* __restrict__ out,
                  float* __restrict__ dump) {
    const int key = blockIdx.x;
    const int off = offsets[key];
    const int cnt = offsets[key + 1] - off;
    if (cnt == 0) return;                       // uniform per block

    const int lane = threadIdx.x & 31;
    const int wave = threadIdx.x >> 5;          // 0..3
    const int nrow = lane & 15;                 // M (for A) / N (for B,D) sub-index
    const int koff = (lane >= 16) ? 2 : 0;      // K split across lane halves
    const int half = (lane >= 16) ? 8 : 0;      // D-matrix row offset
    const int mtiles = (cnt + 15) >> 4;

    const float* __restrict__ Wkey = W + (size_t)key * (D_MODEL * D_IN);
    const float* __restrict__ brow = bias + key * D_MODEL;

    for (int t = 0; t < mtiles; ++t) {          // uniform loop
        // --- A tile: 16 token rows x K=32; float2 per lane per K-step
        int row = t * 16 + nrow;
        int rc  = row < cnt ? row : cnt - 1;    // clamp; garbage rows go to dump
        int tok = list[off + rc];
        const float* __restrict__ erow = emb + (size_t)tok * D_IN + koff;
        v2f a[8];
#pragma unroll
        for (int ks = 0; ks < 8; ++ks) a[ks] = *(const v2f*)(erow + 4 * ks);

        // --- hoist the 8 D-row output pointers out of the N loop
        float* outP[8];
#pragma unroll
        for (int r = 0; r < 8; ++r) {
            int rr  = t * 16 + half + r;
            int rcl = rr < cnt ? rr : cnt - 1;
            int tk  = list[off + rcl];
            int bb  = tk >> 10;                 // / L_DIM
            int ll  = tk & (L_DIM - 1);
            float* real = out + ((size_t)(bb * LP1 + ll + 1)) * D_MODEL;
            outP[r] = (rr < cnt) ? real : dump; // dump: write-only scratch
        }

        for (int n = wave; n < N_TILES; n += 4) {   // uniform per wave
            // --- B tile: W[key][16n+nrow][k], contiguous over k
            const float* __restrict__ wrow = Wkey + (size_t)(n * 16 + nrow) * D_IN + koff;
            v2f bm[8];
#pragma unroll
            for (int ks = 0; ks < 8; ++ks) bm[ks] = *(const v2f*)(wrow + 4 * ks);

            v8f c = {};
#pragma unroll
            for (int ks = 0; ks < 8; ++ks) {
                c = __builtin_amdgcn_wmma_f32_16x16x4_f32(
                        /*neg_a=*/false, a[ks],
                        /*neg_b=*/false, bm[ks],
                        /*c_mod=*/(short)0, c,
                        /*reuse_a=*/false, /*reuse_b=*/false);
            }

            int col  = n * 16 + nrow;
            float bv = brow[col];               // per-N bias
#pragma unroll
            for (int r = 0; r < 8; ++r) {
                outP[r][col] = c[r] + bv;       // branch-free scattered store
            }
        }
    }
}

// ---------------- finalize: tables + CLS row + attn_keep; writes ALL of d_out --

__global__ __launch_bounds__(192)
void te_finalize(const int* __restrict__ pos, const int* __restrict__ sid,
                 const int* __restrict__ mod, const int* __restrict__ role,
                 const unsigned char* __restrict__ mask,
                 const float* __restrict__ cls, const float* __restrict__ pos_tab,
                 const float* __restrict__ id_tab, const float* __restrict__ mod_tab,
                 const float* __restrict__ role_tab, float* __restrict__ out) {
    int rowid = blockIdx.x;                 // 0 .. B*(L+1)-1
    int b = rowid / LP1;
    int r = rowid - b * LP1;
    float* __restrict__ orow = out + (size_t)rowid * D_MODEL;
    float* __restrict__ keep = out + (size_t)B_DIM * LP1 * D_MODEL;
    int col = threadIdx.x * 4;              // 192 threads x float4 = 768

    if (r == 0) {                           // CLS: cls + pos_tab[0] + id_tab[128]
        v4f o = *(const v4f*)(cls + col)
              + *(const v4f*)(pos_tab + col)
              + *(const v4f*)(id_tab + (size_t)NUM_SIGNALS * D_MODEL + col);
        *(v4f*)(orow + col) = o;
        if (threadIdx.x == 0) keep[rowid] = 1.0f;
    } else {
        int t = b * L_DIM + (r - 1);
        int pp = pos[t], ss = sid[t], mm = mod[t], rl = role[t];
        bool keepf = mask[t] != 0;          // uniform per block
        v4f content = {0.f, 0.f, 0.f, 0.f};
        if (keepf) content = *(const v4f*)(orow + col);   // written by te_wmma_gemm
        v4f o = content
              + *(const v4f*)(pos_tab  + (size_t)pp * D_MODEL + col)
              + *(const v4f*)(id_tab   + (size_t)ss * D_MODEL + col)
              + *(const v4f*)(mod_tab  + (size_t)mm * D_MODEL + col)
              + *(const v4f*)(role_tab + (size_t)rl * D_MODEL + col);
        *(v4f*)(orow + col) = o;
        if (threadIdx.x == 0) keep[rowid] = keepf ? 1.0f : 0.0f;
    }
}

// ---------------- launcher ---------------------------------------------------

extern "C" void kernel_launch(void* const* d_in, const int* in_sizes, int n_in,
                              void* d_out, int out_size, void* d_ws, size_t ws_size,
                              hipStream_t stream) {
    const float*         emb      = (const float*)d_in[0];
    const int*           pos      = (const int*)d_in[1];
    const int*           sid      = (const int*)d_in[2];
    const int*           mod      = (const int*)d_in[3];
    const int*           role     = (const int*)d_in[4];
    const unsigned char* mask     = (const unsigned char*)d_in[5];  // jnp.bool_ = 1 byte
    const float*         W        = (const float*)d_in[6];
    const float*         bias     = (const float*)d_in[7];
    const float*         cls      = (const float*)d_in[8];
    const float*         pos_tab  = (const float*)d_in[9];
    const float*         id_tab   = (const float*)d_in[10];
    const float*         mod_tab  = (const float*)d_in[11];
    const float*         role_tab = (const float*)d_in[12];
    float* out = (float*)d_out;

    int*   counts  = (int*)d_ws;                    // NUM_KEYS
    int*   offsets = counts + NUM_KEYS;             // NUM_KEYS + 1
    int*   list    = offsets + NUM_KEYS + 1;        // NTOK
    float* dump    = (float*)(list + NTOK);         // D_MODEL floats, write-only

    te_zero   <<<(NUM_KEYS + 255) / 256, 256, 0, stream>>>(counts);
    te_count  <<<NTOK / 256, 256, 0, stream>>>(sid, role, mask, counts);
    te_scan   <<<1, 512, 0, stream>>>(counts, offsets);
    te_scatter<<<NTOK / 256, 256, 0, stream>>>(sid, role, mask, counts, offsets, list);
    te_wmma_gemm<<<NUM_KEYS, 128, 0, stream>>>(emb, W, bias, offsets, list, out, dump);
    te_finalize<<<B_DIM * LP1, 192, 0, stream>>>(pos, sid, mod, role, mask,
                                                 cls, pos_tab, id_tab, mod_tab, role_tab, out);
}